// InterpBaselineEncoder_32366873543156
// MI455X (gfx1250) — compile-verified
//
#include <hip/hip_runtime.h>

// Problem constants from the reference
#define BB 32
#define HH 128
#define WW 128
#define NTQ 131072      // = 1 << 17
#define DYC 16
#define SPLIT 8         // reduction blocks per batch

typedef float v2f __attribute__((ext_vector_type(2)));
typedef float v4f __attribute__((ext_vector_type(4)));
typedef float v8f __attribute__((ext_vector_type(8)));

// ---------------------------------------------------------------------------
// Kernel 1a: partial sums of yc[b] using WMMA.
// A = all-ones 16x4  =>  D[m,n] += sum_k B[k,n]; every element loaded into B
// appears in exactly one (k,n) slot, so summing one D row over all N equals
// the sum of all loaded elements regardless of the hardware lane layout.
// All 32 lanes of c[0] hold row values (rows 0 and 8, identical), so a
// full-wave lane reduction of c[0] yields 2x the wave total.
// Each of BB*SPLIT blocks reduces a contiguous 32768-float chunk; b128 loads
// + unroll keep several loads in flight (the round-0 version serialized one
// b64 load per WMMA behind s_wait_loadcnt 0).
// ---------------------------------------------------------------------------
__global__ void fill_partial_kernel(const float* __restrict__ yc,
                                    float* __restrict__ partial) {
  const int blk   = blockIdx.x;          // b*SPLIT + s
  const int b     = blk / SPLIT;
  const int s     = blk % SPLIT;
  const int E     = HH * WW * DYC;       // 262144 per batch
  const int chunk = E / SPLIT;           // 32768 per block
  const int lane  = threadIdx.x & 31;
  const int wave  = threadIdx.x >> 5;    // 8 waves per block

  const v4f* __restrict__ src =
      (const v4f*)(yc + (size_t)b * (size_t)E + (size_t)s * (size_t)chunk);

  __shared__ float wsum[8];
  float v;

  const int per_wave = chunk / 8;        // 4096 floats per wave
  const int iters    = per_wave / 128;   // 32 iterations (128 floats each)
  const int idx0     = (wave * per_wave) / 4 + lane;  // in v4f units

#if __has_builtin(__builtin_amdgcn_wmma_f32_16x16x4_f32)
  v8f c = {0.f, 0.f, 0.f, 0.f, 0.f, 0.f, 0.f, 0.f};
  v2f a1;
  a1[0] = 1.0f;
  a1[1] = 1.0f;
#pragma unroll 4
  for (int i = 0; i < iters; ++i) {
    v4f d = src[idx0 + i * 32];          // wave reads 512B contiguous
    v2f b0, b1;
    b0[0] = d[0]; b0[1] = d[1];
    b1[0] = d[2]; b1[1] = d[3];
    c = __builtin_amdgcn_wmma_f32_16x16x4_f32(
        false, a1, false, b0, (short)0, c, false, false);
    c = __builtin_amdgcn_wmma_f32_16x16x4_f32(
        false, a1, false, b1, (short)0, c, false, false);
  }
  v = c[0];
  for (int off = 16; off > 0; off >>= 1) v += __shfl_xor(v, off, 32);
  v *= 0.5f;   // rows 0 and 8 both present across the 32 lanes of c[0]
#else
  v = 0.0f;
#pragma unroll 4
  for (int i = 0; i < iters; ++i) {
    v4f d = src[idx0 + i * 32];
    v += d[0] + d[1] + d[2] + d[3];
  }
  for (int off = 16; off > 0; off >>= 1) v += __shfl_xor(v, off, 32);
#endif

  if (lane == 0) wsum[wave] = v;
  __syncthreads();
  if (threadIdx.x == 0) {
    float t = 0.f;
    for (int i = 0; i < 8; ++i) t += wsum[i];
    partial[blk] = t;
  }
}

// ---------------------------------------------------------------------------
// Kernel 1b: fill[b] = (sum of SPLIT partials) / (H*W*DY). Deterministic
// two-pass reduction (no fp atomics).
// ---------------------------------------------------------------------------
__global__ void fill_finalize_kernel(const float* __restrict__ partial,
                                     float* __restrict__ fill) {
  const int b = threadIdx.x;
  if (b < BB) {
    float t = 0.f;
    for (int i = 0; i < SPLIT; ++i) t += partial[b * SPLIT + i];
    fill[b] = t / (float)(HH * WW * DYC);
  }
}

// ---------------------------------------------------------------------------
// Kernel 2: triangulated bilinear gather-interpolation.
// 4 threads per query point, one float4 DY-chunk each -> coalesced 16B/lane
// non-temporal stores (256 MB output stream must not evict yc from L2).
// Corner gathers use regular-temporal loads so the 32 MB yc table stays
// L2-resident (192 MB L2). global_prefetch_b8 pulls upcoming query coords.
// ---------------------------------------------------------------------------
__global__ void interp_kernel(const float* __restrict__ yc,
                              const float2* __restrict__ xt,
                              const float* __restrict__ fill,
                              v4f* __restrict__ out) {
  const unsigned tid = blockIdx.x * blockDim.x + threadIdx.x;
  const int q        = (int)(tid & 3u);
  const unsigned pt  = tid >> 2;        // flat (b*NT + t)
  const int b        = (int)(pt >> 17); // NTQ == 1<<17

  if (pt + 4096u < (unsigned)(BB * NTQ))
    __builtin_prefetch((const void*)(xt + pt + 4096u), 0, 1);

  const float2 xy = xt[pt];
  const float x = xy.x, y = xy.y;
  const bool inside =
      (x >= 0.f) && (x <= 1.f) && (y >= 0.f) && (y <= 1.f);

  const float u  = x * (float)(WW - 1);
  const float vv = y * (float)(HH - 1);
  const float j0 = fminf(fmaxf(floorf(u), 0.f), (float)(WW - 2));
  const float i0 = fminf(fmaxf(floorf(vv), 0.f), (float)(HH - 2));
  const float fu = u - j0;
  const float fv = vv - i0;

  const int idx00 = (int)i0 * WW + (int)j0;
  const v4f* __restrict__ row =
      (const v4f*)(yc + (size_t)b * (size_t)(HH * WW * DYC));
  const int base = idx00 * (DYC / 4) + q;

  const v4f y00 = row[base];
  const v4f y10 = row[base + (DYC / 4)];
  const v4f y01 = row[base + WW * (DYC / 4)];
  const v4f y11 = row[base + (WW + 1) * (DYC / 4)];

  const v4f lower = y00 + fu * (y10 - y00) + fv * (y01 - y00);
  const v4f upper =
      y11 + (1.f - fu) * (y01 - y11) + (1.f - fv) * (y10 - y11);
  v4f val = (fu + fv <= 1.f) ? lower : upper;

  if (!inside) {
    const float f = fill[b];
    v4f fsplat;
    fsplat[0] = f; fsplat[1] = f; fsplat[2] = f; fsplat[3] = f;
    val = fsplat;
  }

  __builtin_nontemporal_store(val, out + tid);
}

// ---------------------------------------------------------------------------
extern "C" void kernel_launch(void* const* d_in, const int* in_sizes, int n_in,
                              void* d_out, int out_size, void* d_ws,
                              size_t ws_size, hipStream_t stream) {
  // setup_inputs order: xc_off_grid, yc_off_grid, xc_on_grid, yc_on_grid,
  //                     xt, used_modality
  const float*  yc = (const float*)d_in[3];   // [B, H*W, DY]
  const float2* xt = (const float2*)d_in[4];  // [B, NT, 2]

  float* fill    = (float*)d_ws;              // 32 floats
  float* partial = (float*)d_ws + 32;         // BB*SPLIT = 256 floats

  fill_partial_kernel<<<BB * SPLIT, 256, 0, stream>>>(yc, partial);
  fill_finalize_kernel<<<1, 32, 0, stream>>>(partial, fill);

  const int threads = 256;
  const int total   = BB * NTQ * 4;           // 16,777,216 threads
  interp_kernel<<<total / threads, threads, 0, stream>>>(
      yc, xt, fill, (v4f*)d_out);
}